// importance_generator_sr_4518305595840
// MI455X (gfx1250) — compile-verified
//
#include <hip/hip_runtime.h>
#include <math.h>

typedef __attribute__((ext_vector_type(2))) float v2f;
typedef __attribute__((ext_vector_type(8))) float v8f;

#define LN_EPS 1e-5f

__device__ __forceinline__ float sigmoidf_(float x) { return 1.0f / (1.0f + expf(-x)); }

// ---------------- K1: GRU, one block per direction, thread j = gate row j ----
__global__ __launch_bounds__(384) void k_gru(
    const float* __restrict__ x,                       // (4,64)
    const float* __restrict__ w_ih_f, const float* __restrict__ w_hh_f,
    const float* __restrict__ b_ih_f, const float* __restrict__ b_hh_f,
    const float* __restrict__ w_ih_b, const float* __restrict__ w_hh_b,
    const float* __restrict__ b_ih_b, const float* __restrict__ b_hh_b,
    float* __restrict__ hsF, float* __restrict__ hsB)  // (4,128) each
{
    const int dir = blockIdx.x;
    const float* w_ih = dir ? w_ih_b : w_ih_f;
    const float* w_hh = dir ? w_hh_b : w_hh_f;
    const float* b_ih = dir ? b_ih_b : b_ih_f;
    const float* b_hh = dir ? b_hh_b : b_hh_f;
    float* hs = dir ? hsB : hsF;

    __shared__ float h[128];
    __shared__ float gi[384];
    __shared__ float gh[384];

    const int j = threadIdx.x;
    if (j < 128) h[j] = 0.0f;
    __syncthreads();

    for (int kstep = 0; kstep < 4; ++kstep) {
        const int t = dir ? (3 - kstep) : kstep;
        const float* xt = x + t * 64;
        float accI = b_ih[j];
        const float* wi = w_ih + j * 64;
        #pragma unroll 8
        for (int c = 0; c < 64; ++c) accI += wi[c] * xt[c];
        float accH = b_hh[j];
        const float* wh = w_hh + j * 128;
        #pragma unroll 8
        for (int c = 0; c < 128; ++c) accH += wh[c] * h[c];
        gi[j] = accI; gh[j] = accH;
        __syncthreads();
        if (j < 128) {
            float r  = sigmoidf_(gi[j] + gh[j]);
            float z  = sigmoidf_(gi[128 + j] + gh[128 + j]);
            float nn = tanhf(gi[256 + j] + r * gh[256 + j]);
            float hn = (1.0f - z) * nn + z * h[j];
            h[j] = hn;
            hs[t * 128 + j] = hn;
        }
        __syncthreads();
    }
}

// ---------------- K2: LayerNorm + ReLU over the 4 concat rows ----------------
__global__ __launch_bounds__(256) void k_ln(
    const float* __restrict__ hsF, const float* __restrict__ hsB,
    const float* __restrict__ ln_g, const float* __restrict__ ln_b,
    float* __restrict__ hAll)                          // (4,256)
{
    __shared__ float red[256];
    const int tid = threadIdx.x;
    for (int t = 0; t < 4; ++t) {
        float v = (tid < 128) ? hsF[t * 128 + tid] : hsB[t * 128 + (tid - 128)];
        red[tid] = v; __syncthreads();
        for (int s = 128; s > 0; s >>= 1) { if (tid < s) red[tid] += red[tid + s]; __syncthreads(); }
        float mu = red[0] * (1.0f / 256.0f); __syncthreads();
        float d = v - mu;
        red[tid] = d * d; __syncthreads();
        for (int s = 128; s > 0; s >>= 1) { if (tid < s) red[tid] += red[tid + s]; __syncthreads(); }
        float var = red[0] * (1.0f / 256.0f); __syncthreads();
        float y = d * rsqrtf(var + LN_EPS) * ln_g[tid] + ln_b[tid];
        hAll[t * 256 + tid] = fmaxf(y, 0.0f);
    }
}

// ---------------- K3: weight-norm scores via V_WMMA_F32_16X16X4_F32 ----------
// One wave per 16-row tile. A = v rows (fp32), B = h chunk broadcast to all N.
// Row-norm partials accumulated per lane, folded with shfl_xor(16).
__global__ __launch_bounds__(32) void k_scores(
    const float* __restrict__ v0, const float* __restrict__ g0,
    const float* __restrict__ v1, const float* __restrict__ g1,
    const float* __restrict__ v2, const float* __restrict__ g2,
    const float* __restrict__ v3, const float* __restrict__ g3,
    const float* __restrict__ hAll,
    float* __restrict__ sOut)
{
    __shared__ float hl[256];
    __shared__ float dotl[16];
    const int tb = blockIdx.x;
    int layer, tile, eoff;
    const float* vm; const float* gm;
    if (tb < 128)      { layer = 0; tile = tb;       vm = v0; gm = g0; eoff = 0; }
    else if (tb < 384) { layer = 1; tile = tb - 128; vm = v1; gm = g1; eoff = 2048; }
    else if (tb < 640) { layer = 2; tile = tb - 384; vm = v2; gm = g2; eoff = 6144; }
    else               { layer = 3; tile = tb - 640; vm = v3; gm = g3; eoff = 10240; }

    const int lane = threadIdx.x;
    for (int i = lane; i < 256; i += 32) hl[i] = hAll[layer * 256 + i];
    __syncthreads();

    const int lo = lane & 15;       // M row within tile (A), N col (D)
    const int hi = lane >> 4;       // K half-select per ISA layout
    const int R  = tile * 16;
    const float* row = vm + (size_t)(R + lo) * 256 + 2 * hi;
    const float* hb  = hl + 2 * hi;

    v8f acc = {0.f,0.f,0.f,0.f,0.f,0.f,0.f,0.f};
    float pn = 0.0f;
    #pragma unroll 4
    for (int kk = 0; kk < 64; ++kk) {
        v2f a = *(const v2f*)(row + 4 * kk);        // A: K=2*hi, 2*hi+1 of chunk
        v2f b; b.x = hb[4 * kk]; b.y = hb[4 * kk + 1]; // B broadcast across N
        pn += a.x * a.x + a.y * a.y;
        acc = __builtin_amdgcn_wmma_f32_16x16x4_f32(
            false, a, false, b, (short)0, acc, false, false);
    }
    float pn2 = pn + __shfl_xor(pn, 16, 32);        // full ||row||^2 for row lo
    if (lo == 0) {                                  // D: VGPR g -> M = g + 8*hi
        #pragma unroll
        for (int g = 0; g < 8; ++g) dotl[g + 8 * hi] = acc[g];
    }
    __syncthreads();
    if (hi == 0) {
        const int rrow = R + lo;
        float sv = gm[rrow] * dotl[lo] * rsqrtf(pn2);
        sOut[eoff + rrow] = 1.0f / (1.0f + expf(-sv));
    }
}

// ---------------- K4: stable ascending ranks + inverse-order arrays ----------
__global__ __launch_bounds__(256) void k_rank(
    const float* __restrict__ sIn,
    const float* __restrict__ km0, const float* __restrict__ km1,
    const float* __restrict__ km2, const float* __restrict__ km3,
    int* __restrict__ rArr, int* __restrict__ ordArr, float* __restrict__ cArr)
{
    __shared__ float sj[256];
    const int tb = blockIdx.x;
    int bloc, n, eoff; const float* km;
    if (tb < 8)       { bloc = tb;      n = 2048; eoff = 0;     km = km0; }
    else if (tb < 24) { bloc = tb - 8;  n = 4096; eoff = 2048;  km = km1; }
    else if (tb < 40) { bloc = tb - 24; n = 4096; eoff = 6144;  km = km2; }
    else              { bloc = tb - 40; n = 8192; eoff = 10240; km = km3; }
    const int tid = threadIdx.x;
    const int i = bloc * 256 + tid;
    const float si = sIn[eoff + i];
    int cnt = 0;
    for (int j0 = 0; j0 < n; j0 += 256) {
        sj[tid] = sIn[eoff + j0 + tid];
        __syncthreads();
        #pragma unroll 8
        for (int jj = 0; jj < 256; ++jj) {
            const int j = j0 + jj;
            const float v = sj[jj];
            cnt += (v < si) || (v == si && j < i);   // stable ascending rank
        }
        __syncthreads();
    }
    rArr[eoff + i]   = cnt;          // ascending rank of element i
    ordArr[eoff + cnt] = i;          // element with ascending rank cnt
    cArr[eoff + i]   = km[n - 1 - cnt];   // c[j] = k_mask[desc_rank(j)]
}

// ---------------- K5: Gaussian soft-index window + STE value + gather --------
// Wm[i,j] = exp(-(j - r_i)^2): exact fp32 collapse to |j-r_i| <= 10 window.
// out[i] = val[n-1-r_i]  <=>  scatter: out[ord[n-1-m]] = val[m]
__global__ __launch_bounds__(256) void k_mask(
    const int* __restrict__ rArr, const int* __restrict__ ordArr,
    const float* __restrict__ cArr, float* __restrict__ out)
{
    const int tb = blockIdx.x;
    int bloc, n, eoff;
    if (tb < 8)       { bloc = tb;      n = 2048; eoff = 0; }
    else if (tb < 24) { bloc = tb - 8;  n = 4096; eoff = 2048; }
    else if (tb < 40) { bloc = tb - 24; n = 4096; eoff = 6144; }
    else              { bloc = tb - 40; n = 8192; eoff = 10240; }
    const int m  = bloc * 256 + threadIdx.x;
    const int rm = rArr[eoff + m];
    float num = 0.0f, den = 0.0f;
    #pragma unroll
    for (int d = -10; d <= 10; ++d) {
        const int j = rm + d;
        if (j >= 0 && j < n) {
            const float w = expf(-(float)(d * d));
            num += w * cArr[eoff + j];
            den += w;
        }
    }
    const float soft = num / den;
    const float hard = (soft >= 0.5f) ? 1.0f : 0.0f;
    const float val  = 2.0f * soft - hard;   // value of stop_grad(soft-hard)+soft
    const int iout = ordArr[eoff + (n - 1 - m)];
    out[eoff + iout] = val;
}

extern "C" void kernel_launch(void* const* d_in, const int* in_sizes, int n_in,
                              void* d_out, int out_size, void* d_ws, size_t ws_size,
                              hipStream_t stream) {
    const float* gin  = (const float*)d_in[0];
    const float* wihf = (const float*)d_in[1];
    const float* whhf = (const float*)d_in[2];
    const float* bihf = (const float*)d_in[3];
    const float* bhhf = (const float*)d_in[4];
    const float* wihb = (const float*)d_in[5];
    const float* whhb = (const float*)d_in[6];
    const float* bihb = (const float*)d_in[7];
    const float* bhhb = (const float*)d_in[8];
    const float* lng  = (const float*)d_in[9];
    const float* lnb  = (const float*)d_in[10];
    const float* v0 = (const float*)d_in[11]; const float* g0 = (const float*)d_in[12]; const float* km0 = (const float*)d_in[13];
    const float* v1 = (const float*)d_in[14]; const float* g1 = (const float*)d_in[15]; const float* km1 = (const float*)d_in[16];
    const float* v2 = (const float*)d_in[17]; const float* g2 = (const float*)d_in[18]; const float* km2 = (const float*)d_in[19];
    const float* v3 = (const float*)d_in[20]; const float* g3 = (const float*)d_in[21]; const float* km3 = (const float*)d_in[22];

    float* ws   = (float*)d_ws;
    float* hsF  = ws;                    //   512 f
    float* hsB  = ws + 512;              //   512 f
    float* hAll = ws + 1024;             //  1024 f
    float* sArr = ws + 2048;             // 18432 f
    int*   rArr = (int*)(ws + 20480);    // 18432 i
    int*   ordA = (int*)(ws + 38912);    // 18432 i
    float* cArr = ws + 57344;            // 18432 f  (total ~296 KB)
    float* out  = (float*)d_out;         // 18432 f

    k_gru<<<2, 384, 0, stream>>>(gin, wihf, whhf, bihf, bhhf,
                                 wihb, whhb, bihb, bhhb, hsF, hsB);
    k_ln<<<1, 256, 0, stream>>>(hsF, hsB, lng, lnb, hAll);
    k_scores<<<1152, 32, 0, stream>>>(v0, g0, v1, g1, v2, g2, v3, g3, hAll, sArr);
    k_rank<<<72, 256, 0, stream>>>(sArr, km0, km1, km2, km3, rArr, ordA, cArr);
    k_mask<<<72, 256, 0, stream>>>(rArr, ordA, cArr, out);
}